// InitMotionParams_76201309765904
// MI455X (gfx1250) — compile-verified
//
#include <hip/hip_runtime.h>

typedef float v2f __attribute__((ext_vector_type(2)));
typedef float v8f __attribute__((ext_vector_type(8)));

#define KDIM  20   // motion bases
#define KC    5    // K chunks of 4 (WMMA f32 K=4)
#define KP    10   // K pairs (KDIM/2)
#define BT    8    // timesteps
#define NCOLS 72   // BT * 9 (6 rot + 3 transl)
#define NT    5    // N tiles of 16 (NCOLS padded to 80)
#define NPAD  80

__global__ __launch_bounds__(256)
void motion_transform_kernel(const float* __restrict__ motion_rots,    // (K,F,6)
                             const float* __restrict__ motion_transls, // (K,F,3)
                             const float* __restrict__ motion_coefs,   // (G,K)
                             const float* __restrict__ means,          // (G,3)
                             const int*   __restrict__ ts,             // (BT)
                             float*       __restrict__ out,            // (G,BT,3)
                             int G, int F)
{
    // B fragments pre-packed in WMMA-native order:
    // smB[((t*KP + p)*16 + lane)*2 + j] = SM[K=2p+j][col=16t+lane]
    // -> each fragment is one aligned 8-byte LDS load into an even VGPR pair.
    __shared__ float smB[NT * KP * 16 * 2];         // 6.25 KB
    __shared__ float tileBuf[8][16 * NPAD];         // 40 KB, one 16x80 tile per wave

    // ---- build packed B cooperatively (inputs tiny, L2-resident) ----
    for (int idx = threadIdx.x; idx < NT * KP * 16 * 2; idx += 256) {
        int j  = idx & 1;           // component within K-pair
        int lm_ = (idx >> 1) & 15;  // column within N-tile
        int p  = (idx >> 5) % KP;   // K-pair
        int t  = idx / (KP * 16 * 2);
        int k  = 2 * p + j;
        int col = 16 * t + lm_;
        float v = 0.0f;
        if (col < NCOLS) {
            int n = col / 9, jj = col % 9;
            int tt = ts[n];
            if (jj < 6) v = motion_rots[((size_t)k * F + tt) * 6 + jj];
            else        v = motion_transls[((size_t)k * F + tt) * 3 + (jj - 6)];
        }
        smB[idx] = v;
    }
    __syncthreads();

    const int wave = threadIdx.x >> 5;
    const int lane = threadIdx.x & 31;
    const int lm   = lane & 15;        // row (A/C) / column (B/C) index
    const int grp  = lane >> 4;        // half-wave group
    const int tile = blockIdx.x * 8 + wave;
    const int g0   = tile * 16;
    const bool active = (g0 < G);      // wave-uniform

    if (active) {
        // ---- 16 coef rows in A-fragment layout + softmax over K=20 ----
        // lane (row=lm, grp): chunk c holds K = 4c + 2*grp + {0,1}
        int grow = min(g0 + lm, G - 1);
        const float* crow = motion_coefs + (size_t)grow * KDIM;
        float e[2 * KC];
        float mx = -3.0e38f;
        #pragma unroll
        for (int c = 0; c < KC; ++c) {
            v2f x = __builtin_nontemporal_load((const v2f*)(crow + 4 * c + 2 * grp));
            e[2 * c] = x.x; e[2 * c + 1] = x.y;
            mx = fmaxf(mx, fmaxf(x.x, x.y));
        }
        mx = fmaxf(mx, __shfl_xor(mx, 16, 32));              // combine row halves
        float s = 0.0f;
        #pragma unroll
        for (int i = 0; i < 2 * KC; ++i) { e[i] = __expf(e[i] - mx); s += e[i]; }
        s += __shfl_xor(s, 16, 32);
        float inv = 1.0f / s;

        v2f a[KC];
        #pragma unroll
        for (int c = 0; c < KC; ++c) { a[c].x = e[2 * c] * inv; a[c].y = e[2 * c + 1] * inv; }

        // ---- GEMM: 16x80 tile = coefs(16x20) @ SM(20x80), 25 WMMA ops ----
        // c-outer / t-inner: A operand held constant across 5 consecutive WMMAs.
        v8f acc[NT] = {};
        #pragma unroll
        for (int c = 0; c < KC; ++c) {
            int p = 2 * c + grp;   // this half-wave's K-pair for chunk c
            #pragma unroll
            for (int t = 0; t < NT; ++t) {
                v2f b = *(const v2f*)&smB[((t * KP + p) * 16 + lm) * 2];
                acc[t] = __builtin_amdgcn_wmma_f32_16x16x4_f32(
                    /*neg_a=*/false, a[c], /*neg_b=*/false, b,
                    /*c_mod=*/(short)0, acc[t], /*reuse_a=*/false, /*reuse_b=*/false);
            }
        }
        // C layout: VGPR v -> row v + 8*grp ; lane lm -> column 16t+lm
        float* buf = &tileBuf[wave][0];
        #pragma unroll
        for (int t = 0; t < NT; ++t)
            #pragma unroll
            for (int v = 0; v < 8; ++v)
                buf[(v + 8 * grp) * NPAD + 16 * t + lm] = acc[t][v];
    }
    __syncthreads();

    if (active) {
        // ---- per (row, timestep) Gram-Schmidt + affine transform ----
        const float* buf = &tileBuf[wave][0];
        #pragma unroll
        for (int q = 0; q < 4; ++q) {
            int task = lane + 32 * q;      // 0..127
            int row  = task >> 3;          // 0..15
            int n    = task & 7;           // 0..7
            int gidx = g0 + row;
            int gld  = min(gidx, G - 1);

            const float* r9 = buf + row * NPAD + n * 9;
            float a1x = r9[0], a1y = r9[1], a1z = r9[2];
            float a2x = r9[3], a2y = r9[4], a2z = r9[5];
            float tx  = r9[6], ty  = r9[7], tz  = r9[8];

            float mex = means[(size_t)gld * 3 + 0];
            float mey = means[(size_t)gld * 3 + 1];
            float mez = means[(size_t)gld * 3 + 2];

            float n1 = fmaxf(sqrtf(a1x * a1x + a1y * a1y + a1z * a1z), 1e-12f);
            float i1 = 1.0f / n1;
            float b1x = a1x * i1, b1y = a1y * i1, b1z = a1z * i1;
            float d = b1x * a2x + b1y * a2y + b1z * a2z;
            float px = a2x - d * b1x, py = a2y - d * b1y, pz = a2z - d * b1z;
            float n2 = fmaxf(sqrtf(px * px + py * py + pz * pz), 1e-12f);
            float i2 = 1.0f / n2;
            float b2x = px * i2, b2y = py * i2, b2z = pz * i2;
            float b3x = b1y * b2z - b1z * b2y;
            float b3y = b1z * b2x - b1x * b2z;
            float b3z = b1x * b2y - b1y * b2x;

            float ox = b1x * mex + b2x * mey + b3x * mez + tx;
            float oy = b1y * mex + b2y * mey + b3y * mez + ty;
            float oz = b1z * mex + b2z * mey + b3z * mez + tz;

            if (gidx < G) {
                size_t o = ((size_t)gidx * BT + n) * 3;
                __builtin_nontemporal_store(ox, &out[o + 0]);
                __builtin_nontemporal_store(oy, &out[o + 1]);
                __builtin_nontemporal_store(oz, &out[o + 2]);
            }
        }
    }
}

extern "C" void kernel_launch(void* const* d_in, const int* in_sizes, int n_in,
                              void* d_out, int out_size, void* d_ws, size_t ws_size,
                              hipStream_t stream)
{
    const float* motion_rots    = (const float*)d_in[0]; // (K,F,6)
    const float* motion_transls = (const float*)d_in[1]; // (K,F,3)
    const float* motion_coefs   = (const float*)d_in[2]; // (G,K)
    const float* means          = (const float*)d_in[3]; // (G,3)
    const int*   ts             = (const int*)d_in[4];   // (B,)

    int G = in_sizes[3] / 3;
    int F = in_sizes[0] / (6 * KDIM);

    int tiles  = (G + 15) / 16;
    int blocks = (tiles + 7) / 8;   // 8 waves (16-gaussian tiles) per block
    motion_transform_kernel<<<blocks, 256, 0, stream>>>(
        motion_rots, motion_transls, motion_coefs, means, ts,
        (float*)d_out, G, F);
}